// VectorQuantizer2_23175643530036
// MI455X (gfx1250) — compile-verified
//
#include <hip/hip_runtime.h>

// ---------------------------------------------------------------------------
// VQ-VAE vector quantizer for MI455X (gfx1250, wave32, WMMA).
//
//   z   : [16, 8, 32, 32] f32    (z_flat[n][k] = z[n>>10][k][.][n&1023])
//   emb : [16384, 8] f32
//   out : z_q [16,8,32,32] (131072 f32) | loss (1 f32) | idx (16384 f32)
//
// Core: argmin_e (||e||^2 - 2 z.e)  ==  argmin_e (0.5||e||^2 - z.e)
// computed with V_WMMA_F32_16X16X4_F32, A = -z tile (16x8, two K=4 chunks),
// B = emb tile (8x16), C seeded with 0.5||e||^2 so D is the score directly.
// ---------------------------------------------------------------------------

#define N_PTS  16384
#define NE     16384
#define E_DIM  8
#define HW     1024          // 32*32 spatial positions per batch image
#define BETA   0.25f

typedef __attribute__((ext_vector_type(2))) float v2f;
typedef __attribute__((ext_vector_type(8))) float v8f;

// ---- kernel 0: enorm_half[e] = 0.5 * ||emb[e]||^2 ---------------------------
__global__ __launch_bounds__(256) void vq_prep(const float* __restrict__ emb,
                                               float* __restrict__ enorm_half) {
  int e = blockIdx.x * 256 + threadIdx.x;          // 64 blocks * 256 = 16384
  const float* p = emb + e * E_DIM;
  float s = 0.f;
#pragma unroll
  for (int k = 0; k < E_DIM; ++k) s += p[k] * p[k];
  enorm_half[e] = 0.5f * s;
}

// ---- kernel 1: fused distance-GEMM + argmin (WMMA f32 16x16x4) --------------
// One wave per 16-row tile of z_flat; sweeps all 16384 codes in 16-code tiles.
// f32 WMMA layouts (ISA 7.12.2):
//   A 16x4 : lane%16 = M; lane/16 selects K-pair {0,1}/{2,3} in 2 VGPRs
//   B 4x16 : lane%16 = N; lane/16 selects K-pair {0,1}/{2,3} in 2 VGPRs
//   C/D    : lane%16 = N; VGPR v -> M = v + 8*(lane/16)
__global__ __launch_bounds__(256) void vq_argmin(const float* __restrict__ z,
                                                 const float* __restrict__ emb,
                                                 const float* __restrict__ enorm_half,
                                                 int*   __restrict__ idx_i,
                                                 float* __restrict__ idx_f) {
  const int lane  = threadIdx.x & 31;
  const int wave  = threadIdx.x >> 5;
  const int mtile = blockIdx.x * 8 + wave;         // 128 blocks * 8 waves = 1024 tiles
  const int m     = lane & 15;                     // row-in-tile (A) / col-in-tile (B,D)
  const int kb    = (lane >> 4) << 1;              // K base for this lane half: 0 or 2

  // A tile: rows mtile*16 .. +15 of z_flat, negated (score = 0.5||e||^2 - z.e).
  const int   row = mtile * 16 + m;
  const float* zp = z + (row >> 10) * (E_DIM * HW) + (row & (HW - 1));
  v2f a_lo, a_hi;                                  // K = {kb,kb+1} and {kb+4,kb+5}
  a_lo.x = -zp[(kb + 0) * HW];
  a_lo.y = -zp[(kb + 1) * HW];
  a_hi.x = -zp[(kb + 4) * HW];
  a_hi.y = -zp[(kb + 5) * HW];

  float best[8];
  int   bidx[8];
#pragma unroll
  for (int v = 0; v < 8; ++v) { best[v] = 3.4e38f; bidx[v] = 0; }

#pragma unroll 2
  for (int e0 = 0; e0 < NE; e0 += 16) {
    const int    ecol = e0 + m;                    // this lane's code column
    const float* ep   = emb + ecol * E_DIM + kb;   // emb row-major, 8 f32/row
    v2f b_lo = *(const v2f*)(ep);                  // K = kb, kb+1
    v2f b_hi = *(const v2f*)(ep + 4);              // K = kb+4, kb+5
    float en = enorm_half[ecol];
    v8f c = { en, en, en, en, en, en, en, en };    // seed C with 0.5||e||^2
    c = __builtin_amdgcn_wmma_f32_16x16x4_f32(false, a_lo, false, b_lo,
                                              (short)0, c, false, false);
    c = __builtin_amdgcn_wmma_f32_16x16x4_f32(false, a_hi, false, b_hi,
                                              (short)0, c, false, false);
#pragma unroll
    for (int v = 0; v < 8; ++v) {
      if (c[v] < best[v]) { best[v] = c[v]; bidx[v] = ecol; }  // keeps lowest idx
    }
  }

  // Cross-lane argmin: row (v + 8*(lane/16)) lives in the lane's 16-lane half,
  // one candidate per lane. 4-step butterfly; ties -> lower code index.
#pragma unroll
  for (int v = 0; v < 8; ++v) {
    float bv = best[v];
    int   bi = bidx[v];
#pragma unroll
    for (int off = 8; off >= 1; off >>= 1) {
      float ov = __shfl_xor(bv, off, 32);
      int   oi = __shfl_xor(bi, off, 32);
      if (ov < bv || (ov == bv && oi < bi)) { bv = ov; bi = oi; }
    }
    if ((lane & 15) == 0) {
      int grow = mtile * 16 + v + 8 * (lane >> 4);
      idx_i[grow] = bi;
      idx_f[grow] = (float)bi;
    }
  }
}

// ---- kernel 2: gather z_q = emb[idx] back to [b,c,h,w] + partial SSE --------
__global__ __launch_bounds__(256) void vq_gather_loss(const float* __restrict__ z,
                                                      const float* __restrict__ emb,
                                                      const int*   __restrict__ idx_i,
                                                      float* __restrict__ zq,
                                                      float* __restrict__ partial) {
  __shared__ float red[8];
  int n = blockIdx.x * 256 + threadIdx.x;          // 64 blocks * 256 = 16384
  const int    e  = idx_i[n];
  const float* q  = emb + e * E_DIM;
  const int    bo = (n >> 10) * (E_DIM * HW) + (n & (HW - 1));
  float s = 0.f;
#pragma unroll
  for (int d = 0; d < E_DIM; ++d) {
    float qv = q[d];
    float zv = z[bo + d * HW];
    zq[bo + d * HW] = qv;                          // straight-through fwd == z_q
    float df = qv - zv;
    s += df * df;
  }
#pragma unroll
  for (int off = 16; off >= 1; off >>= 1) s += __shfl_xor(s, off, 32);
  if ((threadIdx.x & 31) == 0) red[threadIdx.x >> 5] = s;
  __syncthreads();
  if (threadIdx.x == 0) {
    float t = 0.f;
#pragma unroll
    for (int i = 0; i < 8; ++i) t += red[i];
    partial[blockIdx.x] = t;                       // deterministic (no FP atomics)
  }
}

// ---- kernel 3: loss = (1+beta) * SSE / 131072 -------------------------------
__global__ void vq_finalize(const float* __restrict__ partial,
                            float* __restrict__ loss) {
  int t = threadIdx.x;                             // 32 threads, 64 partials
  float s = partial[t] + partial[t + 32];
#pragma unroll
  for (int off = 16; off >= 1; off >>= 1) s += __shfl_xor(s, off, 32);
  if (t == 0) *loss = s * ((1.f + BETA) / (float)(N_PTS * E_DIM));
}

// ---------------------------------------------------------------------------
extern "C" void kernel_launch(void* const* d_in, const int* in_sizes, int n_in,
                              void* d_out, int out_size, void* d_ws, size_t ws_size,
                              hipStream_t stream) {
  (void)in_sizes; (void)n_in; (void)out_size; (void)ws_size;
  const float* z   = (const float*)d_in[0];        // [16,8,32,32]
  const float* emb = (const float*)d_in[1];        // [16384,8]

  float* out   = (float*)d_out;
  float* zq    = out;                              // 131072 f32
  float* loss  = out + 131072;                     // 1 f32
  float* idxf  = out + 131073;                     // 16384 f32

  float* enorm   = (float*)d_ws;                               // 16384 f32
  int*   idxi    = (int*)((char*)d_ws + NE * sizeof(float));   // 16384 i32
  float* partial = (float*)((char*)d_ws + 2 * NE * sizeof(float)); // 64 f32

  vq_prep       <<<NE / 256, 256, 0, stream>>>(emb, enorm);
  vq_argmin     <<<128,      256, 0, stream>>>(z, emb, enorm, idxi, idxf);
  vq_gather_loss<<<N_PTS/256,256, 0, stream>>>(z, emb, idxi, zq, partial);
  vq_finalize   <<<1,         32, 0, stream>>>(partial, loss);
}